// EnhancedSegMaskLoss_18829136626349
// MI455X (gfx1250) — compile-verified
//
#include <hip/hip_runtime.h>
#include <cstdint>

// ---------------- problem constants ----------------
#define B_      32
#define H_      1024
#define W_      1024
#define ROWS    32                 // rows per block strip
#define STRIPS  (H_ / ROWS)        // 32
#define WPR     (W_ / 32)          // 32 uint32 words per row
#define HALO    3
#define RH      (ROWS + 2 * HALO)  // 38 rows incl. halo
#define NPX     (ROWS * W_)        // 32768 pixels per strip
#define CHUNK   2048               // pixels staged per async chunk (8 KB)
#define NCHUNK  (NPX / CHUNK)      // 16

// weight table: 1 - tanh(d/3) for d=1,2,3 ; 0 for d=4 (outside band)
#define W1C 0.6784870f
#define W2C 0.4172170f
#define W3C 0.2384058f

typedef int v4i __attribute__((ext_vector_type(4)));
typedef __attribute__((address_space(1))) v4i* gptr_v4i;   // global int4*
typedef __attribute__((address_space(3))) v4i* lptr_v4i;   // LDS int4*

// ---------------- CDNA5 helpers ----------------
__device__ __forceinline__ unsigned int ballot32(bool p) {
#if __has_builtin(__builtin_amdgcn_ballot_w32)
    return __builtin_amdgcn_ballot_w32(p);
#else
    return (unsigned int)__ballot(p);
#endif
}

// async copy of 16 bytes global -> LDS (tracked by ASYNCcnt)
__device__ __forceinline__ void async_copy_b128(const float* gp, float* lp) {
#if __has_builtin(__builtin_amdgcn_global_load_async_to_lds_b128)
    __builtin_amdgcn_global_load_async_to_lds_b128(
        (gptr_v4i)(v4i*)gp,     // C-style cast drops const, then addrspacecast to AS1
        (lptr_v4i)(v4i*)lp,     // addrspacecast generic->AS3
        0, 0);
#else
    unsigned int lo = (unsigned int)(unsigned long long)lp;  // LDS byte address
    asm volatile("global_load_async_to_lds_b128 %0, %1, off"
                 :: "v"(lo), "v"(gp) : "memory");
#endif
}

__device__ __forceinline__ void wait_async_le2() {
#if __has_builtin(__builtin_amdgcn_s_wait_asynccnt)
    __builtin_amdgcn_s_wait_asynccnt(2);
#else
    asm volatile("s_wait_asynccnt 2" ::: "memory");
#endif
    asm volatile("" ::: "memory");
}
__device__ __forceinline__ void wait_async_le0() {
#if __has_builtin(__builtin_amdgcn_s_wait_asynccnt)
    __builtin_amdgcn_s_wait_asynccnt(0);
#else
    asm volatile("s_wait_asynccnt 0" ::: "memory");
#endif
    asm volatile("" ::: "memory");
}

__device__ __forceinline__ float wave_red_sum(float v) {
#pragma unroll
    for (int m = 16; m > 0; m >>= 1) v += __shfl_xor(v, m, 32);
    return v;
}

// horizontal dilation helper: OR over x-u..x+u columns using word + neighbors
__device__ __forceinline__ uint32_t hdil_step(uint32_t acc, uint32_t m,
                                              uint32_t ml, uint32_t mr, int u) {
    return acc | (m << u) | (m >> u) | (ml >> (32 - u)) | (mr << (32 - u));
}

// ---------------- main strip kernel ----------------
__global__ __launch_bounds__(256)
void seg_loss_strip_kernel(const float* __restrict__ pred,
                           const float* __restrict__ tgt,
                           float* __restrict__ ws) {
    __shared__ uint32_t fgb[RH][WPR];   // foreground bits (halo rows zeroed)
    __shared__ uint32_t bgb[RH][WPR];   // background bits (halo rows zeroed)
    __shared__ uint32_t h1f[RH][WPR], h2f[RH][WPR], h3f[RH][WPR];
    __shared__ uint32_t h1g[RH][WPR], h2g[RH][WPR], h3g[RH][WPR];
    __shared__ uint32_t p0b[ROWS][WPR], p1b[ROWS][WPR]; // (dist-1) bitplanes
    __shared__ __align__(16) float stage[2][CHUNK];     // async pred double-buffer
    __shared__ float accW[8][8];                        // per-wave partials

    const int tid = threadIdx.x;
    const int wv  = tid >> 5;
    const int ln  = tid & 31;
    const int b   = blockIdx.x / STRIPS;
    const int s   = blockIdx.x % STRIPS;
    const int y0  = s * ROWS;

    // ---- Phase A: ballot bit-pack target strip (+halo) ----
    for (int task = wv; task < RH * WPR; task += 8) {
        const int r  = task >> 5;
        const int w  = task & 31;
        const int gy = y0 - HALO + r;
        const bool valid = (gy >= 0) && (gy < H_);
        float t = 0.0f;
        if (valid) t = tgt[((size_t)b * H_ + gy) * W_ + w * 32 + ln];
        const uint32_t fg = ballot32(valid && (t > 0.5f));
        const uint32_t bg = ballot32(valid && (t <= 0.5f));
        if (ln == 0) { fgb[r][w] = fg; bgb[r][w] = bg; }
    }
    __syncthreads();

    // ---- Phase B1: horizontal dilations d=1..3 per row, fg & bg ----
    for (int task = tid; task < RH * WPR; task += 256) {
        const int r = task >> 5;
        const int w = task & 31;
        {
            uint32_t m  = fgb[r][w];
            uint32_t ml = (w > 0)       ? fgb[r][w - 1] : 0u;
            uint32_t mr = (w < WPR - 1) ? fgb[r][w + 1] : 0u;
            uint32_t a1 = hdil_step(m, m, ml, mr, 1);
            uint32_t a2 = hdil_step(a1, m, ml, mr, 2);
            uint32_t a3 = hdil_step(a2, m, ml, mr, 3);
            h1f[r][w] = a1; h2f[r][w] = a2; h3f[r][w] = a3;
        }
        {
            uint32_t m  = bgb[r][w];
            uint32_t ml = (w > 0)       ? bgb[r][w - 1] : 0u;
            uint32_t mr = (w < WPR - 1) ? bgb[r][w + 1] : 0u;
            uint32_t a1 = hdil_step(m, m, ml, mr, 1);
            uint32_t a2 = hdil_step(a1, m, ml, mr, 2);
            uint32_t a3 = hdil_step(a2, m, ml, mr, 3);
            h1g[r][w] = a1; h2g[r][w] = a2; h3g[r][w] = a3;
        }
    }
    __syncthreads();

    // ---- Phase B2: vertical OR -> per-pixel clipped Chebyshev distance code ----
    for (int task = tid; task < ROWS * WPR; task += 256) {
        const int ro = task >> 5;
        const int w  = task & 31;
        const int r  = ro + HALO;
        uint32_t f1 = h1f[r-1][w] | h1f[r][w] | h1f[r+1][w];
        uint32_t f2 = h2f[r-2][w] | h2f[r-1][w] | h2f[r][w] | h2f[r+1][w] | h2f[r+2][w];
        uint32_t f3 = h3f[r-3][w] | h3f[r-2][w] | h3f[r-1][w] | h3f[r][w]
                    | h3f[r+1][w] | h3f[r+2][w] | h3f[r+3][w];
        uint32_t g1 = h1g[r-1][w] | h1g[r][w] | h1g[r+1][w];
        uint32_t g2 = h2g[r-2][w] | h2g[r-1][w] | h2g[r][w] | h2g[r+1][w] | h2g[r+2][w];
        uint32_t g3 = h3g[r-3][w] | h3g[r-2][w] | h3g[r-1][w] | h3g[r][w]
                    | h3g[r+1][w] | h3g[r+2][w] | h3g[r+3][w];
        const uint32_t fg = fgb[r][w];
        // fg pixels use distance-to-background (g*), bg pixels distance-to-foreground (f*)
        const uint32_t c1 = (fg & g1) | (~fg & f1);
        const uint32_t c2 = (fg & g2) | (~fg & f2);
        const uint32_t c3 = (fg & g3) | (~fg & f3);
        p0b[ro][w] = ~c1 & (c2 | ~c3);  // bit0 of (dist-1)
        p1b[ro][w] = ~c2;               // bit1 of (dist-1)  (c1 subset of c2)
    }
    __syncthreads();

    // ---- Phase C: stream pred through async LDS double-buffer, accumulate ----
    const float* predStrip = pred + ((size_t)b * H_ + y0) * W_;

    auto issue = [&](int c) {
        const int buf  = c & 1;
        const int base = c * CHUNK + wv * 256;
#pragma unroll
        for (int k = 0; k < 2; ++k) {
            const int off = k * 128 + ln * 4;                 // 4 floats / lane / issue
            async_copy_b128(predStrip + base + off, &stage[buf][wv * 256 + off]);
        }
    };

    float s_p = 0.f, s_pt = 0.f, s_t = 0.f, s_bw = 0.f, s_w = 0.f, s_ce = 0.f, s_fo = 0.f;

    issue(0);
    for (int c = 0; c < NCHUNK; ++c) {
        if (c + 1 < NCHUNK) { issue(c + 1); wait_async_le2(); }
        else                { wait_async_le0(); }
        const int buf = c & 1;
#pragma unroll
        for (int k = 0; k < 8; ++k) {
            const int local = wv * 256 + k * 32 + ln;  // wave reads only its own region
            const int px    = c * CHUNK + local;
            const float xv  = stage[buf][local];
            const int ro    = px >> 10;
            const int x     = px & (W_ - 1);
            const int w     = x >> 5;
            const int bit   = x & 31;
            const uint32_t fg = (fgb[ro + HALO][w] >> bit) & 1u;
            const uint32_t b0 = (p0b[ro][w] >> bit) & 1u;
            const uint32_t b1 = (p1b[ro][w] >> bit) & 1u;
            const float wgt = b1 ? (b0 ? 0.0f : W3C) : (b0 ? W2C : W1C);

            const float t   = (float)fg;
            const float ax  = fabsf(xv);
            const float e   = __expf(-ax);
            const float sp1 = 1.0f + e;
            const float bce = fmaxf(xv, 0.0f) - xv * t + __logf(sp1);
            const float rr  = 1.0f / sp1;
            const float pr  = (xv >= 0.0f) ? rr : e * rr;   // sigmoid(xv)
            const float ptv = fg ? pr : (1.0f - pr);
            const float om  = 1.0f - ptv;
            const float al  = fg ? 0.25f : 0.75f;

            s_ce += bce;
            s_fo += al * bce * om * om;
            s_p  += pr;
            s_pt += fg ? pr : 0.0f;
            s_t  += t;
            s_bw += bce * wgt;
            s_w  += wgt;
        }
    }

    // ---- deterministic block reduction (no atomics) ----
    s_p = wave_red_sum(s_p);   s_pt = wave_red_sum(s_pt); s_t  = wave_red_sum(s_t);
    s_bw = wave_red_sum(s_bw); s_w  = wave_red_sum(s_w);
    s_ce = wave_red_sum(s_ce); s_fo = wave_red_sum(s_fo);
    if (ln == 0) {
        accW[wv][0] = s_p;  accW[wv][1] = s_pt; accW[wv][2] = s_t;
        accW[wv][3] = s_bw; accW[wv][4] = s_w;  accW[wv][5] = s_ce; accW[wv][6] = s_fo;
    }
    __syncthreads();
    if (tid < 7) {
        float v = 0.f;
#pragma unroll
        for (int q = 0; q < 8; ++q) v += accW[q][tid];
        ws[(size_t)blockIdx.x * 8 + tid] = v;
    }
}

// ---------------- finalize kernel (1 wave) ----------------
__global__ __launch_bounds__(32)
void seg_loss_final_kernel(const float* __restrict__ ws,
                           const float* __restrict__ logv,
                           float* __restrict__ out) {
    const int b = threadIdx.x;  // one batch per lane
    float Sp = 0.f, Spt = 0.f, St = 0.f, Sbw = 0.f, Sw = 0.f, Sce = 0.f, Sfo = 0.f;
    for (int s = 0; s < STRIPS; ++s) {
        const float* p = ws + ((size_t)(b * STRIPS + s)) * 8;
        Sp += p[0]; Spt += p[1]; St += p[2]; Sbw += p[3]; Sw += p[4]; Sce += p[5]; Sfo += p[6];
    }
    const float dice_b = 1.0f - (2.0f * Spt + 1.0f) / (Sp + St + 1.0f);
    const float iou_b  = 1.0f - (Spt + 1e-6f) / (Sp + St - Spt + 1e-6f);
    const float bnd_b  = Sbw / fmaxf(Sw, 1.0f);

    const float rdice = wave_red_sum(dice_b);
    const float riou  = wave_red_sum(iou_b);
    const float rbnd  = wave_red_sum(bnd_b);
    const float rce   = wave_red_sum(Sce);
    const float rfo   = wave_red_sum(Sfo);

    if (b == 0) {
        const float invN = 1.0f / ((float)B_ * (float)H_ * (float)W_);
        float L[5];
        L[0] = rce * invN;              // ce            * 1.0
        L[1] = rdice / (float)B_;       // dice          * 1.0
        L[2] = 0.5f * rfo * invN;       // focal         * 0.5
        L[3] = 0.3f * riou / (float)B_; // iou           * 0.3
        L[4] = 0.2f * rbnd / (float)B_; // boundary      * 0.2
        float rw[5], sum = 0.f;
#pragma unroll
        for (int i = 0; i < 5; ++i) {
            const float lv = logv[i] * (1.0f / 1.5f);
            rw[i] = 1.0f / (1.0f + __expf(-lv));  // sigmoid(log_vars / TEMP)
            sum += rw[i];
        }
        float tot = 0.f;
#pragma unroll
        for (int i = 0; i < 5; ++i) tot += (rw[i] / sum * 5.0f) * L[i];
        out[0] = tot;
    }
}

// ---------------- launcher ----------------
extern "C" void kernel_launch(void* const* d_in, const int* in_sizes, int n_in,
                              void* d_out, int out_size, void* d_ws, size_t ws_size,
                              hipStream_t stream) {
    const float* pred = (const float*)d_in[0];
    const float* tgt  = (const float*)d_in[1];
    const float* logv = (const float*)d_in[2];
    float* ws  = (float*)d_ws;   // needs B_*STRIPS*8 floats = 32 KB
    float* out = (float*)d_out;

    seg_loss_strip_kernel<<<B_ * STRIPS, 256, 0, stream>>>(pred, tgt, ws);
    seg_loss_final_kernel<<<1, 32, 0, stream>>>(ws, logv, out);
}